// JKNet_6828998001541
// MI455X (gfx1250) — compile-verified
//
#include <hip/hip_runtime.h>
#include <hip/hip_bf16.h>

// ---------------- CDNA5 WMMA types ----------------
typedef float v2f __attribute__((ext_vector_type(2)));
typedef float v8f __attribute__((ext_vector_type(8)));

#define NN      100000
#define EE      1600000
#define F_IN    128
#define HH      64
#define OUTC    40
#define OUTP    48        // lin2 width padded to 3 WMMA tiles
#define NLAYER  3
#define BN_EPS  1e-5f

// ---------------------------------------------------------------------------
// fp32 WMMA GEMM strip kernel: one wave computes a 16-row x (NT*16)-col strip.
//   C[row0:row0+16, 0:ncols_store) = act(A[16,K] @ W[K, NT*16] + bias)
// K is a compile-time multiple of 4 -> fully unrolled V_WMMA_F32_16X16X4_F32
// chain. All B loads unconditional (W pre-padded to NT*16 cols); only the
// store is column-guarded, so EXEC stays all-ones through every WMMA.
// ---------------------------------------------------------------------------
template <int K, int NT>
__global__ __launch_bounds__(32) void wmma_gemm_f32(
    const float* __restrict__ A, int lda,
    const float* __restrict__ W, int ldw,
    const float* __restrict__ bias,
    float* __restrict__ C, int ldc,
    int ncols_store, int do_relu)
{
    const int lane = threadIdx.x;     // 0..31
    const int half = lane >> 4;       // 0: lanes 0-15, 1: lanes 16-31
    const int l16  = lane & 15;
    const int row0 = blockIdx.x * 16;

    v8f acc[NT] = {};

    // A fragment base: lane row = row0+l16; K-pair = k + half*2 (+0,+1)
    const float* Ap = A + (row0 + l16) * lda + half * 2;
    // B fragment base: lane col = l16 (+16*t); K-pair rows = k + half*2 (+0,+1)
    const float* Wp = W + (half * 2) * ldw + l16;

    #pragma unroll
    for (int k = 0; k < K; k += 4) {
        v2f af;
        af.x = Ap[k + 0];
        af.y = Ap[k + 1];
        #pragma unroll
        for (int t = 0; t < NT; ++t) {
            v2f bf;
            bf.x = Wp[(k + 0) * ldw + t * 16];
            bf.y = Wp[(k + 1) * ldw + t * 16];
            acc[t] = __builtin_amdgcn_wmma_f32_16x16x4_f32(
                /*neg_a=*/false, af, /*neg_b=*/false, bf,
                /*c_mod=*/(short)0, acc[t], /*reuse_a=*/false, /*reuse_b=*/false);
        }
    }

    // C/D 16x16 f32 layout: VGPR r -> row row0 + r + half*8, col = tile*16 + l16
    const int rbase = row0 + half * 8;
    #pragma unroll
    for (int t = 0; t < NT; ++t) {
        const int ccol = t * 16 + l16;
        if (ccol < ncols_store) {
            const float bv = bias ? bias[ccol] : 0.0f;
            #pragma unroll
            for (int r = 0; r < 8; ++r) {
                float v = acc[t][r] + bv;
                if (do_relu) v = fmaxf(v, 0.0f);
                C[(rbase + r) * ldc + ccol] = v;
            }
        }
    }
}

// ---------------------------------------------------------------------------
__global__ void fill0_k(float* __restrict__ p, long long n) {
    long long i = (long long)blockIdx.x * blockDim.x + threadIdx.x;
    const long long stride = (long long)gridDim.x * blockDim.x;
    for (; i < n; i += stride) p[i] = 0.0f;
}

// Pad lin2_w [64,40] -> [64,48] (zeros beyond col 40), lin2_b [40] -> [48].
__global__ void pad_lin2_k(const float* __restrict__ w, const float* __restrict__ b,
                           float* __restrict__ wp, float* __restrict__ bp) {
    int i = blockIdx.x * blockDim.x + threadIdx.x;
    if (i < HH * OUTP) {
        int r = i / OUTP, c = i % OUTP;
        wp[i] = (c < OUTC) ? w[r * OUTC + c] : 0.0f;
    }
    if (i < OUTP) bp[i] = (i < OUTC) ? b[i] : 0.0f;
}

__global__ void degree_k(const int* __restrict__ dst, float* __restrict__ deg, int E) {
    int e = blockIdx.x * blockDim.x + threadIdx.x;
    if (e < E) atomicAdd(&deg[dst[e]], 1.0f);
}

__global__ void dis_k(float* __restrict__ deg, int N) {
    int i = blockIdx.x * blockDim.x + threadIdx.x;
    if (i < N) deg[i] = rsqrtf(deg[i] + 1.0f);   // in-place: deg -> deg^-0.5
}

// Edge scatter: agg[dst] += h[src] * (dis[src]*dis[dst]).  16 lanes (float4 each) per edge.
__global__ __launch_bounds__(256) void edge_agg_k(
    const int* __restrict__ src, const int* __restrict__ dst,
    const float* __restrict__ dis, const float* __restrict__ h,
    float* __restrict__ agg, int E)
{
    int gid  = blockIdx.x * blockDim.x + threadIdx.x;
    int edge = gid >> 4;
    int q    = gid & 15;
    if (edge >= E) return;
    const int s = src[edge];
    const int d = dst[edge];
    const float nrm = dis[s] * dis[d];
    const float4 v = ((const float4*)(h + s * HH))[q];
    float* ap = agg + d * HH + q * 4;
    atomicAdd(ap + 0, v.x * nrm);
    atomicAdd(ap + 1, v.y * nrm);
    atomicAdd(ap + 2, v.z * nrm);
    atomicAdd(ap + 3, v.w * nrm);
}

// hpre = agg + tmp*dis^2 + b  (in-place into agg), accumulate per-channel sum/sumsq.
__global__ __launch_bounds__(256) void bn_stats_k(
    float* __restrict__ hpre, const float* __restrict__ tmp,
    const float* __restrict__ dis, const float* __restrict__ bias,
    float* __restrict__ stats, int N)
{
    __shared__ float ssum[256];
    __shared__ float ssq[256];
    const int c    = threadIdx.x & (HH - 1);
    const int rgrp = threadIdx.x >> 6;               // 4 rows per block per pass
    const float b  = bias[c];
    float lsum = 0.0f, lsq = 0.0f;
    for (int n = blockIdx.x * 4 + rgrp; n < N; n += gridDim.x * 4) {
        const float d = dis[n];
        const float v = hpre[n * HH + c] + tmp[n * HH + c] * (d * d) + b;
        hpre[n * HH + c] = v;
        lsum += v;
        lsq  += v * v;
    }
    ssum[threadIdx.x] = lsum;
    ssq[threadIdx.x]  = lsq;
    __syncthreads();
    if (rgrp == 0) {
        const float s = ssum[c] + ssum[HH + c] + ssum[2 * HH + c] + ssum[3 * HH + c];
        const float q = ssq[c]  + ssq[HH + c]  + ssq[2 * HH + c]  + ssq[3 * HH + c];
        atomicAdd(&stats[c], s);
        atomicAdd(&stats[HH + c], q);
    }
}

// stats[c] <- scale, stats[64+c] <- shift
__global__ void bn_finalize_k(float* __restrict__ stats,
                              const float* __restrict__ gamma,
                              const float* __restrict__ beta, int N)
{
    const int c = threadIdx.x;
    const float mu  = stats[c] / (float)N;
    const float var = stats[HH + c] / (float)N - mu * mu;
    const float sc  = gamma[c] * rsqrtf(var + BN_EPS);
    stats[c]      = sc;
    stats[HH + c] = beta[c] - mu * sc;
}

// j[n, col_off + c] = relu(hpre[n,c]*scale + shift)
__global__ __launch_bounds__(256) void bn_apply_k(
    const float* __restrict__ hpre, const float* __restrict__ stats,
    float* __restrict__ j, int col_off, int N)
{
    int idx = blockIdx.x * blockDim.x + threadIdx.x;
    int n = idx >> 6;
    int c = idx & (HH - 1);
    if (n < N) {
        float v = hpre[n * HH + c] * stats[c] + stats[HH + c];
        j[n * (NLAYER * HH) + col_off + c] = fmaxf(v, 0.0f);
    }
}

// ---------------------------------------------------------------------------
extern "C" void kernel_launch(void* const* d_in, const int* in_sizes, int n_in,
                              void* d_out, int out_size, void* d_ws, size_t ws_size,
                              hipStream_t stream) {
    (void)in_sizes; (void)n_in; (void)out_size; (void)ws_size;

    const float* x      = (const float*)d_in[0];
    const int*   eidx   = (const int*)  d_in[1];
    const float* W0     = (const float*)d_in[2];
    const float* b0     = (const float*)d_in[3];
    const float* Wl     = (const float*)d_in[4];
    const float* bl     = (const float*)d_in[5];
    const float* gamma  = (const float*)d_in[6];
    const float* beta   = (const float*)d_in[7];
    const float* lin1_w = (const float*)d_in[8];
    const float* lin1_b = (const float*)d_in[9];
    const float* lin2_w = (const float*)d_in[10];
    const float* lin2_b = (const float*)d_in[11];
    float* out = (float*)d_out;

    const int* src = eidx;          // edge_index[0]
    const int* dst = eidx + EE;     // edge_index[1]

    // workspace layout (floats)
    float* ws    = (float*)d_ws;
    float* dis   = ws;                                   // N
    float* tmp   = dis + NN;                             // N*H   (GEMM out / hmid)
    float* agg   = tmp + (size_t)NN * HH;                // N*H   (edge agg -> hpre)
    float* jbuf  = agg + (size_t)NN * HH;                // N*3H  (JK concat)
    float* stats = jbuf + (size_t)NN * NLAYER * HH;      // 128
    float* w2pad = stats + 2 * HH;                       // 64*48
    float* b2pad = w2pad + HH * OUTP;                    // 48

    const int MT = NN / 16;   // 6250 row tiles (N divisible by 16)

    // degrees -> dis ; pad lin2 weights
    fill0_k<<<1024, 256, 0, stream>>>(dis, (long long)NN);
    pad_lin2_k<<<(HH * OUTP + 255) / 256, 256, 0, stream>>>(lin2_w, lin2_b, w2pad, b2pad);
    degree_k<<<(EE + 255) / 256, 256, 0, stream>>>(dst, dis, EE);
    dis_k<<<(NN + 255) / 256, 256, 0, stream>>>(dis, NN);

    for (int i = 0; i < NLAYER; ++i) {
        fill0_k<<<4096, 256, 0, stream>>>(agg, (long long)NN * HH);
        fill0_k<<<1, 128, 0, stream>>>(stats, 2 * HH);

        // tmp = hin @ W  (bias fused later with the self-loop term)
        if (i == 0) {
            wmma_gemm_f32<F_IN, HH / 16><<<MT, 32, 0, stream>>>(
                x, F_IN, W0, HH, nullptr, tmp, HH, HH, 0);
        } else {
            wmma_gemm_f32<HH, HH / 16><<<MT, 32, 0, stream>>>(
                jbuf + (i - 1) * HH, NLAYER * HH, Wl + (i - 1) * HH * HH, HH,
                nullptr, tmp, HH, HH, 0);
        }

        // agg[dst] += tmp[src] * dis[src]*dis[dst]
        edge_agg_k<<<(EE * 16 + 255) / 256, 256, 0, stream>>>(src, dst, dis, tmp, agg, EE);

        // hpre = agg + tmp*dis^2 + b ; batch stats ; normalize+relu into JK buffer
        const float* b = (i == 0) ? b0 : (bl + (i - 1) * HH);
        bn_stats_k<<<512, 256, 0, stream>>>(agg, tmp, dis, b, stats, NN);
        bn_finalize_k<<<1, HH, 0, stream>>>(stats, gamma + i * HH, beta + i * HH, NN);
        bn_apply_k<<<((size_t)NN * HH + 255) / 256, 256, 0, stream>>>(agg, stats, jbuf, i * HH, NN);
    }

    // hmid = relu(j @ lin1_w + lin1_b)  [N,64] into tmp
    wmma_gemm_f32<NLAYER * HH, HH / 16><<<MT, 32, 0, stream>>>(
        jbuf, NLAYER * HH, lin1_w, HH, lin1_b, tmp, HH, HH, 1);

    // out = hmid @ lin2_w + lin2_b  [N,40]; padded-48 weights, store-guarded to 40
    wmma_gemm_f32<HH, OUTP / 16><<<MT, 32, 0, stream>>>(
        tmp, HH, w2pad, OUTP, b2pad, out, OUTC, OUTC, 0);
}